// GCNModel_57157424775340
// MI455X (gfx1250) — compile-verified
//
#include <hip/hip_runtime.h>
#include <hip/hip_bf16.h>

typedef float v2f __attribute__((ext_vector_type(2)));
typedef float v8f __attribute__((ext_vector_type(8)));

#define HIDDIM 128
#define NHEADS 4
#define NEG_SLOPE 0.2f

// ---------- float <-> order-preserving uint key (for atomicMax on floats) ----------
__device__ __forceinline__ unsigned fenc(float f) {
  unsigned u = __float_as_uint(f);
  return (u & 0x80000000u) ? ~u : (u | 0x80000000u);
}
__device__ __forceinline__ float fdec(unsigned k) {
  unsigned u = (k & 0x80000000u) ? (k & 0x7FFFFFFFu) : ~k;
  return __uint_as_float(u);
}
__device__ __forceinline__ float lrelu(float x) { return x > 0.f ? x : x * NEG_SLOPE; }

// ---------- generic zero ----------
__global__ void k_zero_f32(float* __restrict__ p, long n) {
  long i = blockIdx.x * (long)blockDim.x + threadIdx.x;
  if (i < n) p[i] = 0.f;
}

// ---------- attr_m = attr; then scatter mask_token rows ----------
__global__ void k_copy(const float* __restrict__ a, float* __restrict__ b, long n) {
  long i = blockIdx.x * (long)blockDim.x + threadIdx.x;
  if (i < n) b[i] = a[i];
}
__global__ void k_apply_mask(float* __restrict__ h, const int* __restrict__ midx,
                             const float* __restrict__ mtok, long total) {
  long i = blockIdx.x * (long)blockDim.x + threadIdx.x;
  if (i >= total) return;
  int r = midx[i >> 7];
  h[(long)r * HIDDIM + (i & 127)] = mtok[i & 127];
}

// ---------- Y[nrows,128] = X[nrows,128] @ W[128,128] via V_WMMA_F32_16X16X4_F32 ----------
// One wave -> 16 rows x all 128 cols (8 accumulator tiles).
__global__ void k_gemm128_wmma(const float* __restrict__ X, const float* __restrict__ W,
                               float* __restrict__ Y, int nrows) {
  int gwave = (int)((blockIdx.x * (long)blockDim.x + threadIdx.x) >> 5);
  int lane  = threadIdx.x & 31;
  int row0  = gwave * 16;
  if (row0 >= nrows) return;                 // wave-uniform: EXEC stays all-ones
  int mloc  = lane & 15;                     // M (and N) position within tile
  int khalf = (lane >> 4) << 1;              // lanes 16-31 hold K+2,K+3
  const float* xrow = X + (long)(row0 + mloc) * HIDDIM;
  v8f acc[8] = {};
#pragma unroll 4
  for (int k0 = 0; k0 < HIDDIM; k0 += 4) {
    v2f a;
    a.x = xrow[k0 + khalf];
    a.y = xrow[k0 + khalf + 1];
    const float* wp = W + (long)(k0 + khalf) * HIDDIM + mloc;
#pragma unroll
    for (int t = 0; t < 8; ++t) {
      v2f b;
      b.x = wp[t * 16];
      b.y = wp[HIDDIM + t * 16];
      acc[t] = __builtin_amdgcn_wmma_f32_16x16x4_f32(false, a, false, b,
                                                     (short)0, acc[t], false, false);
    }
  }
  int rseg = (lane >> 4) << 3;               // lanes 16-31 hold rows M=8..15
#pragma unroll
  for (int t = 0; t < 8; ++t) {
    float* yp = Y + (long)(row0 + rseg) * HIDDIM + t * 16 + mloc;
#pragma unroll
    for (int v = 0; v < 8; ++v) yp[(long)v * HIDDIM] = acc[t][v];
  }
}

// ---------- el/er: per-node per-head dot(f[n,h,:], al/ar[h,:]) ----------
__global__ void k_attn_scores(const float* __restrict__ f, const float* __restrict__ al,
                              const float* __restrict__ ar, float* __restrict__ el,
                              float* __restrict__ er, int n) {
  int node = blockIdx.x;
  if (node >= n) return;
  int t = threadIdx.x;            // 0..127 ; head = t>>5, wave32 == one head
  int lane = t & 31;
  float v = f[(long)node * HIDDIM + t];
  float a = v * al[t];
  float b = v * ar[t];
#pragma unroll
  for (int off = 16; off > 0; off >>= 1) {
    a += __shfl_xor(a, off, 32);
    b += __shfl_xor(b, off, 32);
  }
  if (lane == 0) {
    el[node * NHEADS + (t >> 5)] = a;
    er[node * NHEADS + (t >> 5)] = b;
  }
}

// ---------- edge pass 1: e = leaky_relu(el[src]+er[dst]); atomic segment max ----------
__global__ void k_edge_max(const int* __restrict__ src, const int* __restrict__ dst,
                           const float* __restrict__ el, const float* __restrict__ er,
                           float* __restrict__ ebuf, unsigned* __restrict__ mkey, long E) {
  long e = blockIdx.x * (long)blockDim.x + threadIdx.x;
  if (e >= E) return;
  int s = src[e], d = dst[e];
  float4 l = ((const float4*)el)[s];
  float4 r = ((const float4*)er)[d];
  float4 v;
  v.x = lrelu(l.x + r.x); v.y = lrelu(l.y + r.y);
  v.z = lrelu(l.z + r.z); v.w = lrelu(l.w + r.w);
  ((float4*)ebuf)[e] = v;
  unsigned* mk = mkey + (long)d * NHEADS;
  atomicMax(mk + 0, fenc(v.x));
  atomicMax(mk + 1, fenc(v.y));
  atomicMax(mk + 2, fenc(v.z));
  atomicMax(mk + 3, fenc(v.w));
}

// ---------- edge pass 2: ex = exp(e - m[dst]); atomic segment sum ----------
__global__ void k_edge_exp(const int* __restrict__ dst, float* __restrict__ ebuf,
                           const unsigned* __restrict__ mkey, float* __restrict__ ssum, long E) {
  long e = blockIdx.x * (long)blockDim.x + threadIdx.x;
  if (e >= E) return;
  int d = dst[e];
  float4 v = ((float4*)ebuf)[e];
  const unsigned* mk = mkey + (long)d * NHEADS;
  float4 ex;
  ex.x = __expf(v.x - fdec(mk[0]));
  ex.y = __expf(v.y - fdec(mk[1]));
  ex.z = __expf(v.z - fdec(mk[2]));
  ex.w = __expf(v.w - fdec(mk[3]));
  ((float4*)ebuf)[e] = ex;
  float* sp = ssum + (long)d * NHEADS;
  atomicAdd(sp + 0, ex.x);
  atomicAdd(sp + 1, ex.y);
  atomicAdd(sp + 2, ex.z);
  atomicAdd(sp + 3, ex.w);
}

// ---------- edge pass 3: out[dst] += f[src] * (ex / s[dst]) ----------
__global__ void k_edge_aggregate(const int* __restrict__ src, const int* __restrict__ dst,
                                 const float* __restrict__ ebuf, const float* __restrict__ ssum,
                                 const float* __restrict__ f, float* __restrict__ acc, long total) {
  long i = blockIdx.x * (long)blockDim.x + threadIdx.x;
  if (i >= total) return;
  long e = i >> 7;
  int t = (int)(i & 127);
  int h = t >> 5;
  int sn = src[e], dn = dst[e];
  float a = ebuf[e * NHEADS + h] / ssum[(long)dn * NHEADS + h];
  atomicAdd(&acc[(long)dn * HIDDIM + t], f[(long)sn * HIDDIM + t] * a);
}

// ---------- out = relu(acc + b) ----------
__global__ void k_bias_relu(float* __restrict__ h, const float* __restrict__ b, long total) {
  long i = blockIdx.x * (long)blockDim.x + threadIdx.x;
  if (i >= total) return;
  float v = h[i] + b[i & 127];
  h[i] = v > 0.f ? v : 0.f;
}

// ---------- decoder + MSE over masked rows (gathered WMMA GEMM, fused loss) ----------
__global__ void k_decoder_loss(const float* __restrict__ H, const float* __restrict__ Wd,
                               const float* __restrict__ bd, const float* __restrict__ attr,
                               const int* __restrict__ midx, int nmask, float* __restrict__ sum) {
  int gwave = (int)((blockIdx.x * (long)blockDim.x + threadIdx.x) >> 5);
  int lane  = threadIdx.x & 31;
  int row0  = gwave * 16;
  if (row0 >= nmask) return;
  int mloc  = lane & 15;
  int khalf = (lane >> 4) << 1;
  int arow  = row0 + mloc;
  int anode = (arow < nmask) ? midx[arow] : midx[0];
  const float* xrow = H + (long)anode * HIDDIM;
  v8f acc[8] = {};
#pragma unroll 4
  for (int k0 = 0; k0 < HIDDIM; k0 += 4) {
    v2f a;
    a.x = xrow[k0 + khalf];
    a.y = xrow[k0 + khalf + 1];
    const float* wp = Wd + (long)(k0 + khalf) * HIDDIM + mloc;
#pragma unroll
    for (int t = 0; t < 8; ++t) {
      v2f b;
      b.x = wp[t * 16];
      b.y = wp[HIDDIM + t * 16];
      acc[t] = __builtin_amdgcn_wmma_f32_16x16x4_f32(false, a, false, b,
                                                     (short)0, acc[t], false, false);
    }
  }
  int rseg = (lane >> 4) << 3;
  float lsum = 0.f;
#pragma unroll
  for (int t = 0; t < 8; ++t) {
    int col = t * 16 + mloc;
#pragma unroll
    for (int v = 0; v < 8; ++v) {
      int r = row0 + rseg + v;
      if (r < nmask) {
        int node = midx[r];
        float diff = acc[t][v] + bd[col] - attr[(long)node * HIDDIM + col];
        lsum += diff * diff;
      }
    }
  }
#pragma unroll
  for (int off = 16; off > 0; off >>= 1) lsum += __shfl_xor(lsum, off, 32);
  if (lane == 0) atomicAdd(sum, lsum);
}

__global__ void k_finalize(const float* __restrict__ sum, float* __restrict__ out, int nmask) {
  out[0] = sum[0] / ((float)nmask * (float)HIDDIM);
}

// =====================================================================

static inline unsigned cdiv(long a, long b) { return (unsigned)((a + b - 1) / b); }

static void run_gat_layer(const float* hin, const int* src, const int* dst,
                          const float* W, const float* al, const float* ar, const float* b,
                          float* fbuf, float* hout, float* el, float* er,
                          unsigned* mkey, float* ssum, float* ebuf,
                          int n, long E, hipStream_t stream) {
  long nh = (long)n * HIDDIM;
  // zero accumulators: hout [N,128], mkey+ssum (adjacent [N,4] each)
  k_zero_f32<<<cdiv(nh, 256), 256, 0, stream>>>(hout, nh);
  k_zero_f32<<<cdiv((long)n * NHEADS * 2, 256), 256, 0, stream>>>((float*)mkey, (long)n * NHEADS * 2);
  // f = hin @ W
  long waves = (n + 15) / 16;
  k_gemm128_wmma<<<cdiv(waves * 32, 256), 256, 0, stream>>>(hin, W, fbuf, n);
  // attention scores
  k_attn_scores<<<(unsigned)n, HIDDIM, 0, stream>>>(fbuf, al, ar, el, er, n);
  // edge softmax (3 passes) + aggregation
  k_edge_max<<<cdiv(E, 256), 256, 0, stream>>>(src, dst, el, er, ebuf, mkey, E);
  k_edge_exp<<<cdiv(E, 256), 256, 0, stream>>>(dst, ebuf, mkey, ssum, E);
  k_edge_aggregate<<<cdiv(E * HIDDIM, 256), 256, 0, stream>>>(src, dst, ebuf, ssum, fbuf, hout, E * HIDDIM);
  // bias + relu
  k_bias_relu<<<cdiv(nh, 256), 256, 0, stream>>>(hout, b, nh);
}

extern "C" void kernel_launch(void* const* d_in, const int* in_sizes, int n_in,
                              void* d_out, int out_size, void* d_ws, size_t ws_size,
                              hipStream_t stream) {
  const float* attr = (const float*)d_in[0];
  const int*   src  = (const int*)d_in[1];
  const int*   dst  = (const int*)d_in[2];
  const int*   midx = (const int*)d_in[3];
  const float* W0   = (const float*)d_in[4];
  const float* al0  = (const float*)d_in[5];
  const float* ar0  = (const float*)d_in[6];
  const float* b0   = (const float*)d_in[7];
  const float* W1   = (const float*)d_in[8];
  const float* al1  = (const float*)d_in[9];
  const float* ar1  = (const float*)d_in[10];
  const float* b1   = (const float*)d_in[11];
  const float* Wd   = (const float*)d_in[12];
  const float* bd   = (const float*)d_in[13];
  const float* mtok = (const float*)d_in[14];

  int  n     = in_sizes[0] / HIDDIM;   // IN_DIM == HID == 128
  long E     = in_sizes[1];
  int  nmask = in_sizes[3];
  long nh    = (long)n * HIDDIM;

  // workspace layout
  char* ws = (char*)d_ws;
  float*    buf0 = (float*)ws;                 ws += nh * sizeof(float);   // h_in / layer1 out
  float*    buf1 = (float*)ws;                 ws += nh * sizeof(float);   // f
  float*    buf2 = (float*)ws;                 ws += nh * sizeof(float);   // layer0 out
  float*    el   = (float*)ws;                 ws += (long)n * NHEADS * sizeof(float);
  float*    er   = (float*)ws;                 ws += (long)n * NHEADS * sizeof(float);
  unsigned* mkey = (unsigned*)ws;              ws += (long)n * NHEADS * sizeof(unsigned);
  float*    ssum = (float*)ws;                 ws += (long)n * NHEADS * sizeof(float);
  float*    ebuf = (float*)ws;                 ws += E * NHEADS * sizeof(float);
  float*    lsum = (float*)ws;                 ws += sizeof(float);

  // 1) masked features
  k_copy<<<cdiv(nh, 256), 256, 0, stream>>>(attr, buf0, nh);
  k_apply_mask<<<cdiv((long)nmask * HIDDIM, 256), 256, 0, stream>>>(buf0, midx, mtok,
                                                                    (long)nmask * HIDDIM);
  // 2) GAT layer 0: buf0 -> buf2
  run_gat_layer(buf0, src, dst, W0, al0, ar0, b0, buf1, buf2, el, er, mkey, ssum, ebuf,
                n, E, stream);
  // 3) GAT layer 1: buf2 -> buf0
  run_gat_layer(buf2, src, dst, W1, al1, ar1, b1, buf1, buf0, el, er, mkey, ssum, ebuf,
                n, E, stream);
  // 4) decoder + masked MSE
  k_zero_f32<<<1, 32, 0, stream>>>(lsum, 1);
  long dwaves = (nmask + 15) / 16;
  k_decoder_loss<<<cdiv(dwaves * 32, 256), 256, 0, stream>>>(buf0, Wd, bd, attr, midx,
                                                             nmask, lsum);
  k_finalize<<<1, 1, 0, stream>>>(lsum, (float*)d_out, nmask);
}